// AdaptiveSpectralConvolution_72181220377216
// MI455X (gfx1250) — compile-verified
//
#include <hip/hip_runtime.h>
#include <hip/hip_bf16.h>

#define C_   128
#define H_   256
#define W_   256
#define B_   4
#define WF_  129
#define NB_  8
#define BLK_ 16
#define LAM_ 0.5f

typedef __attribute__((ext_vector_type(2))) float v2f;
typedef __attribute__((ext_vector_type(8))) float v8f;

// ---- CDNA5 async global->LDS path (guarded; falls back to load+ds_store) ----
#if defined(__gfx1250__) && __has_builtin(__builtin_amdgcn_global_load_async_to_lds_b128) && \
    __has_builtin(__builtin_amdgcn_global_load_async_to_lds_b32)
#define HAS_ASYNC_LDS 1
typedef int vi4 __attribute__((vector_size(16)));            // int4 (gcc vector form, as in builtin sig)
typedef __attribute__((address_space(1))) vi4 GAvi4;         // global int4*
typedef __attribute__((address_space(3))) vi4 LAvi4;         // LDS int4*
typedef __attribute__((address_space(1))) int GAi;           // global int*
typedef __attribute__((address_space(3))) int LAi;           // LDS int*

__device__ __forceinline__ void async_g2l_b128(const float* g, float* l) {
  __builtin_amdgcn_global_load_async_to_lds_b128((GAvi4*)g, (LAvi4*)l, 0, 0);
}
__device__ __forceinline__ void async_g2l_b32(const float* g, float* l) {
  __builtin_amdgcn_global_load_async_to_lds_b32((GAi*)g, (LAi*)l, 0, 0);
}
__device__ __forceinline__ void async_wait0() {
#if __has_builtin(__builtin_amdgcn_s_wait_asynccnt)
  __builtin_amdgcn_s_wait_asynccnt(0);
#else
  asm volatile("s_wait_asynccnt 0x0" ::: "memory");
#endif
}
#endif

__device__ __forceinline__ v8f wmma4(v2f a, v2f b, v8f c) {
  // V_WMMA_F32_16X16X4_F32 : D = A(16x4) * B(4x16) + C(16x16)
  return __builtin_amdgcn_wmma_f32_16x16x4_f32(false, a, false, b, (short)0, c,
                                               false, false);
}

__device__ __forceinline__ v8f vzero() {
  v8f z;
#pragma unroll
  for (int e = 0; e < 8; e++) z[e] = 0.f;
  return z;
}

// ---------------------------------------------------------------------------
// Kernel 1: bias = x @ conv_w^T + conv_b   (M=262144, N=128, K=128, f32 WMMA)
// ---------------------------------------------------------------------------
__global__ __launch_bounds__(256) void bias_gemm(const float* __restrict__ x,
                                                 const float* __restrict__ cw,
                                                 const float* __restrict__ cb,
                                                 float* __restrict__ out) {
  __shared__ float xs[64 * 132];  // 64-row slab, pitch 132 (bank spread, 16B ok)
  const int t = threadIdx.x;
  const long gr0 = (long)blockIdx.x * 64;

  // keep conv_w hot (global_prefetch_b8); 64KB stays L2/WGP$-resident
  __builtin_prefetch(cw + t * 64, 0, 1);

  // cooperative coalesced staging of 64x128 x-slab into LDS
#ifdef HAS_ASYNC_LDS
#pragma unroll
  for (int i = 0; i < 8; i++) {
    int idx4 = t + 256 * i;          // float4 index
    int row  = idx4 >> 5;            // 32 float4 per row
    int col  = (idx4 & 31) * 4;
    async_g2l_b128(x + (gr0 + row) * C_ + col, xs + row * 132 + col);
  }
  async_wait0();
#else
#pragma unroll
  for (int i = 0; i < 8; i++) {
    int idx4 = t + 256 * i;
    int row  = idx4 >> 5;
    int col  = (idx4 & 31) * 4;
    const float4 v = *(const float4*)(x + (gr0 + row) * C_ + col);
    *(float4*)(xs + row * 132 + col) = v;
  }
#endif
  __syncthreads();

  const int w    = t >> 5;
  const int lane = t & 31;
  const int m    = lane & 15;   // M row (A) and N col (B/D) index
  const int hi   = lane >> 4;
  const int m0   = (w & 3) * 16;       // row tile within slab
  const int nt0  = (w >> 2) * 4;       // 4 N-tiles per wave

  v8f acc[4];
#pragma unroll
  for (int j = 0; j < 4; j++) acc[j] = vzero();

  for (int kc = 0; kc < 32; kc++) {    // K = 128 in chunks of 4
    const int kk = kc * 4 + hi * 2;
    v2f a;
    a.x = xs[(m0 + m) * 132 + kk];
    a.y = xs[(m0 + m) * 132 + kk + 1];
#pragma unroll
    for (int j = 0; j < 4; j++) {
      const int n0 = (nt0 + j) * 16;
      v2f b;  // B[k][n] = conv_w[n][k] (transposed read; 64KB stays cache-hot)
      b.x = cw[(long)(n0 + m) * C_ + kk];
      b.y = cw[(long)(n0 + m) * C_ + kk + 1];
      acc[j] = wmma4(a, b, acc[j]);
    }
  }

#pragma unroll
  for (int j = 0; j < 4; j++) {
    const int col  = (nt0 + j) * 16 + m;
    const float bv = cb[col];
#pragma unroll
    for (int v = 0; v < 8; v++) {
      const long row = gr0 + m0 + v + 8 * hi;  // D: lanes16-31 hold M+8
      out[row * C_ + col] = acc[j][v] + bv;
    }
  }
}

// ---------------------------------------------------------------------------
// 256-point in-place radix-2 DIT FFT in LDS (bit-reversed input order).
// 16 lines per workgroup, pitch 257 floats (conflict-free), 256 threads.
// ---------------------------------------------------------------------------
__device__ __forceinline__ void fft256(float* sre, float* sim, int t,
                                       float sign) {
  const int line = t & 15;
  const int bset = t >> 4;  // 0..15
  float* re = sre + line * 257;
  float* im = sim + line * 257;
  for (int s = 0; s < 8; s++) {
    __syncthreads();
    const int half = 1 << s;
#pragma unroll
    for (int k = 0; k < 8; k++) {
      const int bf  = bset + (k << 4);      // 0..127
      const int grp = bf >> s;
      const int pos = bf & (half - 1);
      const int i0  = (grp << (s + 1)) + pos;
      const int i1  = i0 + half;
      const float ang = sign * 3.14159265358979323846f * (float)pos /
                        (float)half;
      float wi, wr;
      __sincosf(ang, &wi, &wr);
      const float xr = re[i1], xi = im[i1];
      const float tr = xr * wr - xi * wi;
      const float ti = xr * wi + xi * wr;
      const float ur = re[i0], ui = im[i0];
      re[i1] = ur - tr;
      im[i1] = ui - ti;
      re[i0] = ur + tr;
      im[i0] = ui + ti;
    }
  }
  __syncthreads();
}

// ---------------------------------------------------------------------------
// Kernel 2: real FFT along W -> XR/XI [b,h,f(0..128),c]
// ---------------------------------------------------------------------------
__global__ __launch_bounds__(256) void rowfft_fwd(const float* __restrict__ x,
                                                  float* __restrict__ XR,
                                                  float* __restrict__ XI) {
  __shared__ float sre[16 * 257];
  __shared__ float sim[16 * 257];
  const int t = threadIdx.x;
  const int line = t & 15, bset = t >> 4;
  const int cq = blockIdx.x & 7;
  const int bh = blockIdx.x >> 3;       // b*H + h
  const int c  = cq * 16 + line;
  const long base = (long)bh * (W_ * C_) + c;

#ifdef HAS_ASYNC_LDS
#pragma unroll
  for (int k = 0; k < 16; k++) {
    const int j  = bset + (k << 4);
    const int jr = __brev((unsigned)j) >> 24;
    async_g2l_b32(x + base + (long)j * C_, sre + line * 257 + jr);
    sim[line * 257 + jr] = 0.f;
  }
  async_wait0();
#else
#pragma unroll
  for (int k = 0; k < 16; k++) {
    const int j  = bset + (k << 4);
    const float v = x[base + (long)j * C_];
    const int jr = __brev((unsigned)j) >> 24;
    sre[line * 257 + jr] = v;
    sim[line * 257 + jr] = 0.f;
  }
#endif
  fft256(sre, sim, t, -1.f);

  const long ob = (long)bh * (WF_ * C_) + c;
  for (int k = 0; k < 9; k++) {
    const int f = bset + (k << 4);
    if (f <= 128) {
      XR[ob + (long)f * C_] = sre[line * 257 + f];
      XI[ob + (long)f * C_] = sim[line * 257 + f];
    }
  }
}

// ---------------------------------------------------------------------------
// Kernel 3/5: complex FFT along H, in place on XR/XI. sign=-1 fwd (+1/256
// ortho scale folded into store), sign=+1 inverse (scale=1).
// ---------------------------------------------------------------------------
__global__ __launch_bounds__(256) void colfft(float* __restrict__ XR,
                                              float* __restrict__ XI,
                                              float sign, float scale) {
  __shared__ float sre[16 * 257];
  __shared__ float sim[16 * 257];
  const int t = threadIdx.x;
  const int line = t & 15, bset = t >> 4;
  const int cq = blockIdx.x & 7;
  const int bf = blockIdx.x >> 3;
  const int b  = bf / WF_;
  const int f  = bf % WF_;
  const int c  = cq * 16 + line;
  const long base = ((long)b * H_ * WF_ + f) * C_ + c;
  const long hstr = (long)WF_ * C_;

#ifdef HAS_ASYNC_LDS
#pragma unroll
  for (int k = 0; k < 16; k++) {
    const int j  = bset + (k << 4);
    const int jr = __brev((unsigned)j) >> 24;
    async_g2l_b32(XR + base + (long)j * hstr, sre + line * 257 + jr);
    async_g2l_b32(XI + base + (long)j * hstr, sim + line * 257 + jr);
  }
  async_wait0();
#else
#pragma unroll
  for (int k = 0; k < 16; k++) {
    const int j  = bset + (k << 4);
    const int jr = __brev((unsigned)j) >> 24;
    sre[line * 257 + jr] = XR[base + (long)j * hstr];
    sim[line * 257 + jr] = XI[base + (long)j * hstr];
  }
#endif
  fft256(sre, sim, t, sign);

#pragma unroll
  for (int k = 0; k < 16; k++) {
    const int j = bset + (k << 4);
    XR[base + (long)j * hstr] = sre[line * 257 + j] * scale;
    XI[base + (long)j * hstr] = sim[line * 257 + j] * scale;
  }
}

// ---------------------------------------------------------------------------
// Kernel 4: per-frequency block-diagonal complex MLP, f32 WMMA, in place.
// One wave = 16 spectral points; loops 8 channel-blocks.
//   r1 = relu(xr*W10 - xi*W11 + b1[0]); i1 = relu(xr*W11 - xi*W10 + b1[1])
//   r2 = r1*W20 - i1*W21 + b2[0];       i2 = r2*W21 - i1*W20 + b2[1]
//   out = softshrink(r2, i2)
// ---------------------------------------------------------------------------
__global__ __launch_bounds__(256) void spectral_mlp(
    float* __restrict__ XR, float* __restrict__ XI,
    const float* __restrict__ w1f, const float* __restrict__ b1f,
    const float* __restrict__ w2f, const float* __restrict__ b2f) {
  __shared__ float sA[8][16 * 17];  // per-wave D->A re-layout scratch
  __shared__ float sB[8][16 * 17];
  const int t = threadIdx.x;
  const int w = t >> 5, lane = t & 31;
  const int n = lane & 15, hi = lane >> 4;
  float* ldsA = sA[w];
  float* ldsB = sB[w];
  const long p0 = ((long)blockIdx.x * 8 + w) * 16;  // 16-point tile base
  const float b1r = b1f[n], b1i = b1f[16 + n];
  const float b2r = b2f[n], b2i = b2f[16 + n];

  for (int blk = 0; blk < NB_; blk++) {
    const int cbch = blk * BLK_;
    const float* W10 = w1f + (0 * NB_ + blk) * 256;
    const float* W11 = w1f + (1 * NB_ + blk) * 256;
    const float* W20 = w2f + (0 * NB_ + blk) * 256;
    const float* W21 = w2f + (1 * NB_ + blk) * 256;

    // ---- stage 1: four 16x16x16 GEMMs ----
    v8f aRr = vzero(), aRi = vzero(), aIr = vzero(), aIi = vzero();
#pragma unroll
    for (int s = 0; s < 4; s++) {
      const int kk = s * 4 + hi * 2;
      const long ra = (p0 + n) * C_ + cbch + kk;  // A row m = lane&15
      v2f ar, ai, bb0, bb1;
      ar.x = XR[ra];     ar.y = XR[ra + 1];
      ai.x = XI[ra];     ai.y = XI[ra + 1];
      bb0.x = W10[kk * 16 + n];  bb0.y = W10[(kk + 1) * 16 + n];
      bb1.x = W11[kk * 16 + n];  bb1.y = W11[(kk + 1) * 16 + n];
      aRr = wmma4(ar, bb0, aRr);
      aRi = wmma4(ai, bb1, aRi);
      aIr = wmma4(ar, bb1, aIr);
      aIi = wmma4(ai, bb0, aIi);
    }
#pragma unroll
    for (int v = 0; v < 8; v++) {
      const int M = v + 8 * hi;
      ldsA[M * 17 + n] = fmaxf(aRr[v] - aRi[v] + b1r, 0.f);  // r1
      ldsB[M * 17 + n] = fmaxf(aIr[v] - aIi[v] + b1i, 0.f);  // i1
    }
    __syncthreads();

    // ---- stage 2, pass 1: r1*W20, i1*W21, i1*W20 ----
    v8f aA = vzero(), aB = vzero(), aD = vzero();
#pragma unroll
    for (int s = 0; s < 4; s++) {
      const int kk = s * 4 + hi * 2;
      v2f a1, a2, b20, b21;
      a1.x = ldsA[n * 17 + kk];  a1.y = ldsA[n * 17 + kk + 1];
      a2.x = ldsB[n * 17 + kk];  a2.y = ldsB[n * 17 + kk + 1];
      b20.x = W20[kk * 16 + n];  b20.y = W20[(kk + 1) * 16 + n];
      b21.x = W21[kk * 16 + n];  b21.y = W21[(kk + 1) * 16 + n];
      aA = wmma4(a1, b20, aA);
      aB = wmma4(a2, b21, aB);
      aD = wmma4(a2, b20, aD);
    }
    __syncthreads();
#pragma unroll
    for (int v = 0; v < 8; v++) {
      const int M = v + 8 * hi;
      ldsA[M * 17 + n] = aA[v] - aB[v] + b2r;  // r2 (pre-softshrink, feeds i2)
    }
    __syncthreads();

    // ---- stage 2, pass 2: r2*W21 ----
    v8f aC = vzero();
#pragma unroll
    for (int s = 0; s < 4; s++) {
      const int kk = s * 4 + hi * 2;
      v2f a1, b21;
      a1.x = ldsA[n * 17 + kk];  a1.y = ldsA[n * 17 + kk + 1];
      b21.x = W21[kk * 16 + n];  b21.y = W21[(kk + 1) * 16 + n];
      aC = wmma4(a1, b21, aC);
    }

    // softshrink + in-place store
#pragma unroll
    for (int v = 0; v < 8; v++) {
      const int M = v + 8 * hi;
      const float r2 = aA[v] - aB[v] + b2r;
      const float i2 = aC[v] - aD[v] + b2i;
      const float rr = copysignf(fmaxf(fabsf(r2) - LAM_, 0.f), r2);
      const float ii = copysignf(fmaxf(fabsf(i2) - LAM_, 0.f), i2);
      const long gi = (p0 + M) * C_ + cbch + n;
      XR[gi] = rr;
      XI[gi] = ii;
    }
    __syncthreads();
  }
}

// ---------------------------------------------------------------------------
// Kernel 6: Hermitian-extended inverse real FFT along W, accumulate into out
// (which already holds the conv-bias path). Final ortho scale 1/256.
// (Kept on the VALU load path: the conjugate branch must negate imag.)
// ---------------------------------------------------------------------------
__global__ __launch_bounds__(256) void rowifft_add(const float* __restrict__ XR,
                                                   const float* __restrict__ XI,
                                                   float* __restrict__ out) {
  __shared__ float sre[16 * 257];
  __shared__ float sim[16 * 257];
  const int t = threadIdx.x;
  const int line = t & 15, bset = t >> 4;
  const int cq = blockIdx.x & 7;
  const int bh = blockIdx.x >> 3;
  const int c  = cq * 16 + line;
  const long sb = (long)bh * (WF_ * C_) + c;

#pragma unroll
  for (int k = 0; k < 16; k++) {
    const int j = bset + (k << 4);
    float vr, vi;
    if (j <= 128) {
      vr = XR[sb + (long)j * C_];
      vi = XI[sb + (long)j * C_];
    } else {
      const int jm = 256 - j;  // conj symmetry
      vr = XR[sb + (long)jm * C_];
      vi = -XI[sb + (long)jm * C_];
    }
    const int jr = __brev((unsigned)j) >> 24;
    sre[line * 257 + jr] = vr;
    sim[line * 257 + jr] = vi;
  }
  fft256(sre, sim, t, 1.f);

  const long ob = (long)bh * (W_ * C_) + c;
#pragma unroll
  for (int k = 0; k < 16; k++) {
    const int j = bset + (k << 4);
    const long idx = ob + (long)j * C_;
    out[idx] = out[idx] + sre[line * 257 + j] * (1.0f / 256.0f);
  }
}

// ---------------------------------------------------------------------------
extern "C" void kernel_launch(void* const* d_in, const int* in_sizes, int n_in,
                              void* d_out, int out_size, void* d_ws,
                              size_t ws_size, hipStream_t stream) {
  (void)in_sizes; (void)n_in; (void)out_size; (void)ws_size;
  const float* x  = (const float*)d_in[0];
  const float* w1 = (const float*)d_in[1];
  const float* b1 = (const float*)d_in[2];
  const float* w2 = (const float*)d_in[3];
  const float* b2 = (const float*)d_in[4];
  const float* cw = (const float*)d_in[5];
  const float* cb = (const float*)d_in[6];
  float* out = (float*)d_out;

  float* XR = (float*)d_ws;                       // [B,H,WF,C] real plane
  float* XI = XR + (size_t)B_ * H_ * WF_ * C_;    // [B,H,WF,C] imag plane

  bias_gemm  <<<4096, 256, 0, stream>>>(x, cw, cb, out);
  rowfft_fwd <<<8192, 256, 0, stream>>>(x, XR, XI);
  colfft     <<<4128, 256, 0, stream>>>(XR, XI, -1.f, 1.f / 256.f);
  spectral_mlp<<<1032, 256, 0, stream>>>(XR, XI, w1, b1, w2, b2);
  colfft     <<<4128, 256, 0, stream>>>(XR, XI, 1.f, 1.f);
  rowifft_add<<<8192, 256, 0, stream>>>(XR, XI, out);
}